// GeneralLayer_34007551050423
// MI455X (gfx1250) — compile-verified
//
#include <hip/hip_runtime.h>
#include <hip/hip_bf16.h>

// ---------------------------------------------------------------------------
// GCNConv + BatchNorm + ReLU + L2-normalize for MI455X (gfx1250, wave32).
// GEMM uses V_WMMA_F32_16X16X4_F32 (full fp32 matrix pipe). Everything else
// is L2-resident streaming + atomics.
// ---------------------------------------------------------------------------

typedef float v2f __attribute__((ext_vector_type(2)));
typedef float v8f __attribute__((ext_vector_type(8)));

#define DIMF 64

// --- degree init: deg[i] = 1 (self loop) ----------------------------------
__global__ __launch_bounds__(256) void k_deg_init(float* __restrict__ deg, int N) {
    int i = blockIdx.x * blockDim.x + threadIdx.x;
    if (i < N) deg[i] = 1.0f;
}

// --- degree accumulate: deg[dst] += 1 per edge ----------------------------
__global__ __launch_bounds__(256) void k_deg_accum(const long long* __restrict__ ei,
                                                   float* __restrict__ deg, int E) {
    int e = blockIdx.x * blockDim.x + threadIdx.x;
    if (e < E) {
        long long d = ei[(long long)E + e];  // dst row
        atomicAdd(&deg[d], 1.0f);
    }
}

// --- dinv = rsqrt(deg) (in place) -----------------------------------------
__global__ __launch_bounds__(256) void k_deg_rsqrt(float* __restrict__ deg, int N) {
    int i = blockIdx.x * blockDim.x + threadIdx.x;
    if (i < N) deg[i] = rsqrtf(deg[i]);
}

// --- h = x @ W via V_WMMA_F32_16X16X4_F32; out = h * dinv^2 (self loop) ---
// Block = 256 threads = 8 waves; each wave computes 16 rows x 64 cols.
// A 16x4 f32 frag layout: lanes 0-15 hold K={0,1}, lanes 16-31 hold K={2,3}
// (float2 per lane). B 4x16 frag mirrors with N = lane%16. C/D: 8 VGPRs,
// lanes 0-15 -> M=r, lanes 16-31 -> M=8+r, N = lane%16.
__global__ __launch_bounds__(256) void k_gemm_wmma(const float* __restrict__ x,
                                                   const float* __restrict__ W,
                                                   const float* __restrict__ dinv,
                                                   float* __restrict__ h,
                                                   float* __restrict__ out, int N) {
    const int lane  = threadIdx.x & 31;
    const int wave  = threadIdx.x >> 5;                 // 0..7
    const int m0    = blockIdx.x * 128 + wave * 16;     // wave-uniform row base
    if (m0 >= N) return;                                // uniform exit: EXEC stays full

    const int mrow  = lane & 15;
    const int khalf = lane >> 4;                        // 0 or 1
    int row  = m0 + mrow;
    int rowc = row < N ? row : N - 1;                   // clamp loads for tail

    v8f acc0 = {}, acc1 = {}, acc2 = {}, acc3 = {};

    #pragma unroll
    for (int k0 = 0; k0 < DIMF; k0 += 4) {
        // A fragment: x[rowc][k0 + 2*khalf + {0,1}] as one float2
        v2f a = *(const v2f*)(x + (size_t)rowc * DIMF + k0 + 2 * khalf);

        // B fragments for the four 16-wide N tiles; W is L0/L2 resident.
        const int kb = k0 + 2 * khalf;
        v2f b0, b1, b2, b3;
        b0.x = W[(kb + 0) * DIMF + mrow +  0];  b0.y = W[(kb + 1) * DIMF + mrow +  0];
        b1.x = W[(kb + 0) * DIMF + mrow + 16];  b1.y = W[(kb + 1) * DIMF + mrow + 16];
        b2.x = W[(kb + 0) * DIMF + mrow + 32];  b2.y = W[(kb + 1) * DIMF + mrow + 32];
        b3.x = W[(kb + 0) * DIMF + mrow + 48];  b3.y = W[(kb + 1) * DIMF + mrow + 48];

        acc0 = __builtin_amdgcn_wmma_f32_16x16x4_f32(false, a, false, b0, (short)0, acc0, false, false);
        acc1 = __builtin_amdgcn_wmma_f32_16x16x4_f32(false, a, false, b1, (short)0, acc1, false, false);
        acc2 = __builtin_amdgcn_wmma_f32_16x16x4_f32(false, a, false, b2, (short)0, acc2, false, false);
        acc3 = __builtin_amdgcn_wmma_f32_16x16x4_f32(false, a, false, b3, (short)0, acc3, false, false);
    }

    // Store h and out = h * dinv[row]^2
    #pragma unroll
    for (int r = 0; r < 8; ++r) {
        int row2 = m0 + khalf * 8 + r;
        if (row2 < N) {
            float dv = dinv[row2];
            float sw = dv * dv;
            size_t base = (size_t)row2 * DIMF + mrow;
            float v0 = acc0[r], v1 = acc1[r], v2 = acc2[r], v3 = acc3[r];
            h[base +  0] = v0;  out[base +  0] = v0 * sw;
            h[base + 16] = v1;  out[base + 16] = v1 * sw;
            h[base + 32] = v2;  out[base + 32] = v2 * sw;
            h[base + 48] = v3;  out[base + 48] = v3 * sw;
        }
    }
}

// --- edge scatter: out[dst] += dinv[src]*dinv[dst] * h[src], 64 lanes/edge -
__global__ __launch_bounds__(256) void k_scatter(const long long* __restrict__ ei,
                                                 const float* __restrict__ h,
                                                 const float* __restrict__ dinv,
                                                 float* __restrict__ out, int E) {
    long long idx = (long long)blockIdx.x * blockDim.x + threadIdx.x;
    long long e = idx >> 6;
    int d = (int)(idx & 63);
    if (e >= E) return;
    long long s = ei[e];
    long long t = ei[(long long)E + e];
    float norm = dinv[s] * dinv[t];
    atomicAdd(&out[t * DIMF + d], norm * h[s * DIMF + d]);
}

// --- zero the BN stats accumulators (d_ws is not re-poisoned per call) ----
__global__ void k_zero_stats(float* __restrict__ sums, float* __restrict__ sumsq) {
    int i = threadIdx.x;
    if (i < DIMF) { sums[i] = 0.0f; sumsq[i] = 0.0f; }
}

// --- per-column sum / sum of squares over N rows --------------------------
__global__ __launch_bounds__(256) void k_bn_stats(const float* __restrict__ out,
                                                  float* __restrict__ sums,
                                                  float* __restrict__ sumsq, int N) {
    int col = threadIdx.x & 63;
    int rowStart  = (int)((blockIdx.x * blockDim.x + threadIdx.x) >> 6);
    int rowStride = (int)((gridDim.x * blockDim.x) >> 6);
    float s = 0.0f, ss = 0.0f;
    for (int r = rowStart; r < N; r += rowStride) {
        float v = out[(size_t)r * DIMF + col];
        s += v; ss += v * v;
    }
    atomicAdd(&sums[col], s);
    atomicAdd(&sumsq[col], ss);
}

// --- fold BN affine into y = v*scale + shift (bias b cancels in BN) -------
__global__ void k_bn_finalize(const float* __restrict__ sums,
                              const float* __restrict__ sumsq,
                              const float* __restrict__ gamma,
                              const float* __restrict__ beta,
                              float* __restrict__ scale,
                              float* __restrict__ shift, int N) {
    int c = threadIdx.x;
    if (c < DIMF) {
        float invN = 1.0f / (float)N;
        float mean = sums[c] * invN;
        float var  = fmaxf(sumsq[c] * invN - mean * mean, 0.0f);
        float inv  = rsqrtf(var + 1e-5f);
        float sc   = gamma[c] * inv;
        scale[c] = sc;
        shift[c] = beta[c] - mean * sc;
    }
}

// --- BN apply + ReLU + row L2 normalize; one wave32 per row ---------------
__global__ __launch_bounds__(256) void k_row_norm(float* __restrict__ out,
                                                  const float* __restrict__ scale,
                                                  const float* __restrict__ shift, int N) {
    int lane = threadIdx.x & 31;
    long long row = (long long)(blockIdx.x * blockDim.x + threadIdx.x) >> 5;
    if (row >= N) return;
    size_t base = (size_t)row * DIMF;
    float v0 = fmaf(out[base + lane],      scale[lane],      shift[lane]);
    float v1 = fmaf(out[base + lane + 32], scale[lane + 32], shift[lane + 32]);
    v0 = fmaxf(v0, 0.0f);
    v1 = fmaxf(v1, 0.0f);
    float ss = v0 * v0 + v1 * v1;
    #pragma unroll
    for (int m = 16; m >= 1; m >>= 1) ss += __shfl_xor(ss, m, 32);
    float inv = 1.0f / fmaxf(sqrtf(ss), 1e-12f);
    out[base + lane]      = v0 * inv;
    out[base + lane + 32] = v1 * inv;
}

extern "C" void kernel_launch(void* const* d_in, const int* in_sizes, int n_in,
                              void* d_out, int out_size, void* d_ws, size_t ws_size,
                              hipStream_t stream) {
    const float*     x     = (const float*)d_in[0];
    const long long* ei    = (const long long*)d_in[1];   // int64 edge_index [2,E]
    const float*     W     = (const float*)d_in[2];
    // d_in[3] = b : cancels exactly inside BatchNorm -> folded away
    const float*     gamma = (const float*)d_in[4];
    const float*     beta  = (const float*)d_in[5];
    float*           out   = (float*)d_out;

    const int N = in_sizes[0] / DIMF;
    const int E = in_sizes[1] / 2;

    // Workspace layout (all rewritten every call; graph-replay safe)
    float* h     = (float*)d_ws;                 // N*64
    float* deg   = h + (size_t)N * DIMF;         // N (deg -> dinv in place)
    float* sums  = deg + N;                      // 64
    float* sumsq = sums + DIMF;                  // 64
    float* scale = sumsq + DIMF;                 // 64
    float* shift = scale + DIMF;                 // 64

    k_deg_init <<<(N + 255) / 256, 256, 0, stream>>>(deg, N);
    k_deg_accum<<<(E + 255) / 256, 256, 0, stream>>>(ei, deg, E);
    k_deg_rsqrt<<<(N + 255) / 256, 256, 0, stream>>>(deg, N);

    k_gemm_wmma<<<(N + 127) / 128, 256, 0, stream>>>(x, W, deg, h, out, N);

    long long scatterThreads = (long long)E * DIMF;
    k_scatter<<<(unsigned)((scatterThreads + 255) / 256), 256, 0, stream>>>(ei, h, deg, out, E);

    k_zero_stats <<<1, 128, 0, stream>>>(sums, sumsq);
    k_bn_stats   <<<512, 256, 0, stream>>>(out, sums, sumsq, N);
    k_bn_finalize<<<1, 64, 0, stream>>>(sums, sumsq, gamma, beta, scale, shift, N);

    long long rowThreads = (long long)N * 32;
    k_row_norm<<<(unsigned)((rowThreads + 255) / 256), 256, 0, stream>>>(out, scale, shift, N);
}